// GCN__57071525429601
// MI455X (gfx1250) — compile-verified
//
#include <hip/hip_runtime.h>
#include <hip/hip_bf16.h>

typedef float v2f __attribute__((ext_vector_type(2)));
typedef float v8f __attribute__((ext_vector_type(8)));

#define N_NODES  50000
#define N_EDGES  800000
#define N_GRAPHS 128
#define IN_DIM   256
#define HD1      128
#define HD2      64
#define OUT_DIM  10

// ------------------------------------------------ CDNA5 async global->LDS DMA
// 16B per lane, tracked by ASYNCcnt (CDNA5 ISA)
__device__ __forceinline__ void async_g2lds_b128(unsigned lds_byte_off, const void* gaddr) {
    asm volatile("global_load_async_to_lds_b128 %0, %1, off"
                 :: "v"(lds_byte_off), "v"(gaddr) : "memory");
}
__device__ __forceinline__ void wait_asynccnt0() {
    asm volatile("s_wait_asynccnt 0x0" ::: "memory");
}

// ---------------------------------------------------------------- utilities
__global__ __launch_bounds__(256) void fill_f32(float* __restrict__ p, float v, int n) {
    int i = blockIdx.x * blockDim.x + threadIdx.x;
    if (i < n) p[i] = v;
}

__global__ __launch_bounds__(256) void deg_count(const int* __restrict__ col,
                                                 float* __restrict__ deg) {
    int e = blockIdx.x * blockDim.x + threadIdx.x;
    if (e < N_EDGES) atomicAdd(&deg[col[e]], 1.0f);
}

__global__ __launch_bounds__(256) void deg_rsqrt(float* __restrict__ deg) {
    int i = blockIdx.x * blockDim.x + threadIdx.x;
    if (i < N_NODES) {
        float d = deg[i];
        deg[i] = d > 0.0f ? rsqrtf(d) : 0.0f;   // deg >= 1 always (self loop)
    }
}

// ---------------------------------------------------------------- WMMA GEMM
// Block: MT row-tiles of 16 rows, N/16 waves (one 16-wide col-tile per wave).
// Each wave computes MT 16x16 tiles -> B fragments amortized MT times.
// fp32 path: V_WMMA_F32_16X16X4_F32 (exact fp32, matches reference numerics).
template<int K, int N, int MT>
__global__ __launch_bounds__(N * 2) void gemm_wmma_f32(
        const float* __restrict__ A, const float* __restrict__ W,
        float* __restrict__ D) {
    constexpr int LDA  = K + 4;            // 260/132: rows hit distinct LDS banks
    constexpr int ROWS = 16 * MT;
    constexpr int NTHR = N * 2;
    __shared__ float As[ROWS * LDA];

    const int tid  = threadIdx.x;
    const int row0 = blockIdx.x * ROWS;

    // stage ROWS x K strip of A into LDS with async b128 DMA
    for (int i = tid; i < ROWS * K / 4; i += NTHR) {
        const int r  = i / (K / 4);
        const int kq = i % (K / 4);
        const unsigned lds_off = (unsigned)(uintptr_t)(&As[r * LDA + kq * 4]);
        async_g2lds_b128(lds_off, A + (size_t)(row0 + r) * K + kq * 4);
    }
    wait_asynccnt0();
    __syncthreads();

    const int wave = tid >> 5;
    const int lane = tid & 31;
    const int col0 = wave * 16;
    const int half = lane >> 4;            // 0: K-pair {k,k+1}; 1: {k+2,k+3}
    const int l16  = lane & 15;

    v8f acc[MT];
    #pragma unroll
    for (int t = 0; t < MT; ++t) acc[t] = (v8f){};

    #pragma unroll 4
    for (int k = 0; k < K; k += 4) {
        const int kk = k + half * 2;
        v2f b;                              // B frag: N = col0+l16, K = kk..kk+1
        b.x = W[(size_t)kk * N + col0 + l16];
        b.y = W[(size_t)(kk + 1) * N + col0 + l16];
        #pragma unroll
        for (int t = 0; t < MT; ++t) {      // A frag: M = t*16+l16, K = kk..kk+1
            const float* ap = &As[(t * 16 + l16) * LDA + kk];
            v2f a; a.x = ap[0]; a.y = ap[1];
            acc[t] = __builtin_amdgcn_wmma_f32_16x16x4_f32(
                        false, a, false, b, (short)0, acc[t], false, false);
        }
    }

    // D layout: VGPR v -> M = v + 8*half, N = l16
    float* dbase = D + (size_t)row0 * N + col0 + l16;
    #pragma unroll
    for (int t = 0; t < MT; ++t) {
        #pragma unroll
        for (int v = 0; v < 8; ++v) {
            const int m = t * 16 + v + half * 8;
            dbase[(size_t)m * N] = acc[t][v];
        }
    }
}

// ---------------------------------------------------------------- scatter
// msg = h[row] * dinv[row]*dinv[col]; agg[col] += msg  (atomic f32)
template<int F>
__global__ __launch_bounds__(256) void scatter_edges(
        const float* __restrict__ h, const int* __restrict__ row,
        const int* __restrict__ col, const float* __restrict__ dinv,
        float* __restrict__ agg) {
    constexpr int PER = 256 / F;           // edges per block
    const int e = blockIdx.x * PER + (threadIdx.x / F);
    const int j = threadIdx.x & (F - 1);
    if (e < N_EDGES) {
        const int r = row[e];
        const int c = col[e];
        const float nrm = dinv[r] * dinv[c];
        atomicAdd(&agg[(size_t)c * F + j], h[(size_t)r * F + j] * nrm);
    }
}

// agg = relu(agg + h*dinv^2 + b)  -- self-loop folded in, no atomics needed
template<int F>
__global__ __launch_bounds__(256) void bias_relu_selfloop(
        float* __restrict__ agg, const float* __restrict__ h,
        const float* __restrict__ dinv, const float* __restrict__ b) {
    const int idx = blockIdx.x * blockDim.x + threadIdx.x;
    if (idx < N_NODES * F) {
        const int node = idx / F;
        const int j    = idx & (F - 1);
        const float di = dinv[node];
        const float v  = agg[idx] + h[idx] * di * di + b[j];
        agg[idx] = v > 0.0f ? v : 0.0f;
    }
}

// ---------------------------------------------------------------- pooling
// post-ReLU values >= 0 so uint order == float order; pooled init 0
__global__ __launch_bounds__(256) void pool_max(
        const float* __restrict__ a2, const int* __restrict__ batch,
        unsigned* __restrict__ pooled) {
    const int idx = blockIdx.x * blockDim.x + threadIdx.x;
    if (idx < N_NODES * HD2) {
        const int node = idx / HD2;
        const int j    = idx & (HD2 - 1);
        atomicMax(&pooled[batch[node] * HD2 + j], __float_as_uint(a2[idx]));
    }
}

// ---------------------------------------------------------------- final FC
__global__ __launch_bounds__(128) void final_fc(
        const float* __restrict__ g, const float* __restrict__ Wfc,
        const float* __restrict__ bfc, float* __restrict__ out) {
    const int i = blockIdx.x * blockDim.x + threadIdx.x;
    if (i < N_GRAPHS * OUT_DIM) {
        const int gi = i / OUT_DIM;
        const int o  = i % OUT_DIM;
        float acc = bfc[o];
        #pragma unroll 8
        for (int j = 0; j < HD2; ++j)
            acc += g[gi * HD2 + j] * Wfc[j * OUT_DIM + o];
        out[i] = acc;
    }
}

// ---------------------------------------------------------------- driver
extern "C" void kernel_launch(void* const* d_in, const int* in_sizes, int n_in,
                              void* d_out, int out_size, void* d_ws, size_t ws_size,
                              hipStream_t stream) {
    const float* x    = (const float*)d_in[0];
    const int*   ei   = (const int*)d_in[1];
    const int*   row  = ei;                 // edge_index[0]
    const int*   col  = ei + N_EDGES;       // edge_index[1]
    const int*   batch= (const int*)d_in[2];
    const float* W1   = (const float*)d_in[3];
    const float* b1   = (const float*)d_in[4];
    const float* W2   = (const float*)d_in[5];
    const float* b2   = (const float*)d_in[6];
    const float* Wfc  = (const float*)d_in[7];
    const float* bfc  = (const float*)d_in[8];
    float* out = (float*)d_out;

    // workspace carve-up (floats), ~64 MB total
    float* ws     = (float*)d_ws;
    float* dinv   = ws;                          // deg -> rsqrt in place
    float* h1     = dinv + 50048;                // 50000*128
    float* agg1   = h1   + (size_t)N_NODES * HD1;// 50000*128
    float* h2     = h1;                          // alias: h1 dead after epilogue 1
    float* agg2   = agg1 + (size_t)N_NODES * HD1;// 50000*64
    float* pooled = agg2 + (size_t)N_NODES * HD2;// 128*64

    // init: deg=1 (self loop), agg1/agg2/pooled = 0
    fill_f32<<<(N_NODES + 255) / 256, 256, 0, stream>>>(dinv, 1.0f, N_NODES);
    fill_f32<<<(N_NODES * HD1) / 256, 256, 0, stream>>>(agg1, 0.0f, N_NODES * HD1);
    fill_f32<<<(N_NODES * HD2) / 256, 256, 0, stream>>>(agg2, 0.0f, N_NODES * HD2);
    fill_f32<<<(N_GRAPHS * HD2 + 255) / 256, 256, 0, stream>>>(pooled, 0.0f, N_GRAPHS * HD2);

    // normalization: deg over col (+self), dinv = 1/sqrt(deg)
    deg_count<<<N_EDGES / 256, 256, 0, stream>>>(col, dinv);
    deg_rsqrt<<<(N_NODES + 255) / 256, 256, 0, stream>>>(dinv);

    // layer 1: h1 = x @ W1 ; agg1 += scatter ; a1 = relu(agg1 + self + b1)
    gemm_wmma_f32<IN_DIM, HD1, 5><<<N_NODES / 80, HD1 * 2, 0, stream>>>(x, W1, h1);
    scatter_edges<HD1><<<N_EDGES / (256 / HD1), 256, 0, stream>>>(h1, row, col, dinv, agg1);
    bias_relu_selfloop<HD1><<<(N_NODES * HD1) / 256, 256, 0, stream>>>(agg1, h1, dinv, b1);

    // layer 2: h2 = a1 @ W2 ; agg2 += scatter ; a2 = relu(agg2 + self + b2)
    gemm_wmma_f32<HD1, HD2, 5><<<N_NODES / 80, HD2 * 2, 0, stream>>>(agg1, W2, h2);
    scatter_edges<HD2><<<N_EDGES / (256 / HD2), 256, 0, stream>>>(h2, row, col, dinv, agg2);
    bias_relu_selfloop<HD2><<<(N_NODES * HD2) / 256, 256, 0, stream>>>(agg2, h2, dinv, b2);

    // global max pool + FC
    pool_max<<<(N_NODES * HD2) / 256, 256, 0, stream>>>(agg2, batch, (unsigned*)pooled);
    final_fc<<<(N_GRAPHS * OUT_DIM + 127) / 128, 128, 0, stream>>>(pooled, Wfc, bfc, out);
}